// PolarConv2D_15728170238157
// MI455X (gfx1250) — compile-verified
//
#include <hip/hip_runtime.h>

// Depthwise 7x7 conv (VALID) on MI455X via banded-matrix WMMA F32 16x16x4,
// register-resident banded weights, and double-buffered async global->LDS staging.
// x: (B=16, C=256, 128, 128) f32 ; w: (256, 7, 7) f32 ; out: (16,256,122,122) f32

typedef __attribute__((ext_vector_type(2))) float v2f;
typedef __attribute__((ext_vector_type(8))) float v8f;
typedef __attribute__((ext_vector_type(4))) float f4;
typedef int v4i __attribute__((vector_size(4 * sizeof(int))));
typedef __attribute__((address_space(1))) v4i* gv4i_p;   // global-side pointer for async builtin
typedef __attribute__((address_space(3))) v4i* lv4i_p;   // LDS-side pointer for async builtin

#define TILE_ROWS  22          // 16 output rows + 6 halo
#define LDS_STRIDE 132         // padded row stride (floats): even & 16B-aligned, bank-friendly
#define TILE_ELEMS (TILE_ROWS * LDS_STRIDE + 8)  // +8 tail: max dead A read stays in-array (zeros)
#define WPAD_W     40          // per-kh weight row padded so d in [-16,23] -> idx d+16, zeros outside band

#if defined(__has_builtin)
#  if __has_builtin(__builtin_amdgcn_global_load_async_to_lds_b128)
#    define HAVE_ASYNC_LDS 1
#  endif
#endif

__global__ __launch_bounds__(256) void dwconv7x7_wmma(const float* __restrict__ x,
                                                      const float* __restrict__ w,
                                                      float* __restrict__ out,
                                                      int nchan)
{
    constexpr int H = 128, W = 128, HO = 122, WO = 122;

    __shared__ float wpad[7 * WPAD_W];
    __shared__ float tile[2][TILE_ELEMS];    // double buffer for load/compute overlap

    const int plane = blockIdx.x;            // b*C + c
    const int c     = plane % nchan;
    const int tid   = threadIdx.x;
    const int lane  = tid & 31;
    const int wave  = tid >> 5;              // 8 waves, 16 output cols each
    const int n     = lane & 15;             // B/C/D column, also A row (m)
    const int khalf = lane >> 4;             // selects K pair within K=4 chunk
    const int wo0   = wave * 16;

    const float* xp = x   + (size_t)plane * H * W;
    float*       op = out + (size_t)plane * HO * WO;

    // ---- one-time: zero weight band table and both LDS tiles (finite A/B everywhere) ----
    for (int i = tid; i < 7 * WPAD_W; i += 256) wpad[i] = 0.0f;
    for (int i = tid; i < 2 * TILE_ELEMS; i += 256) tile[0][i] = 0.0f;
    __syncthreads();
    if (tid < 49) wpad[(tid / 7) * WPAD_W + 16 + (tid % 7)] = w[c * 49 + tid];
    __syncthreads();

    // ---- banded B matrices, register-resident for the whole plane ----
    // B_kh[k, col] = w[kh, k - col] for 0 <= k-col <= 6, else 0 (read from zero-padded table)
    v2f bk[7][6];
#pragma unroll
    for (int kh = 0; kh < 7; ++kh)
#pragma unroll
        for (int kk = 0; kk < 6; ++kk) {
            const int d0 = 4 * kk + 2 * khalf - n + 16;        // 1..38, in-range of WPAD_W
            bk[kh][kk].x = wpad[kh * WPAD_W + d0];
            bk[kh][kk].y = wpad[kh * WPAD_W + d0 + 1];
        }

    // ---- async stage of one 22x128 input slab into an LDS buffer ----
    auto stage = [&](int ho0, float* buf) {
        const int ravail = (H - ho0 < TILE_ROWS) ? (H - ho0) : TILE_ROWS;
        for (int ch = tid; ch < ravail * 32; ch += 256) {
            const int r = ch >> 5, p = (ch & 31) * 4;
#if defined(HAVE_ASYNC_LDS)
            __builtin_amdgcn_global_load_async_to_lds_b128(
                (gv4i_p)(xp + (size_t)(ho0 + r) * W + p),
                (lv4i_p)&buf[r * LDS_STRIDE + p],
                0, 0);
#else
            *(f4*)&buf[r * LDS_STRIDE + p] = *(const f4*)(xp + (size_t)(ho0 + r) * W + p);
#endif
        }
    };

    stage(0, tile[0]);                       // prologue: prefetch tile 0

    for (int ty = 0; ty < 8; ++ty) {
        const int    ho0 = ty * 16;
        const float* cur = tile[ty & 1];

#if defined(HAVE_ASYNC_LDS)
#  if __has_builtin(__builtin_amdgcn_s_wait_asynccnt)
        __builtin_amdgcn_s_wait_asynccnt(0);
#  else
        asm volatile("s_wait_asynccnt 0" ::: "memory");
#  endif
#endif
        __syncthreads();                     // tile ty landed everywhere; compute ty-1 done everywhere

        if (ty < 7) stage(ho0 + 16, tile[(ty + 1) & 1]);   // overlap next fetch with this compute

        // ---- 42x V_WMMA_F32_16X16X4_F32: Out += A(x slab) x B(banded weights) ----
        v8f acc = {0.f, 0.f, 0.f, 0.f, 0.f, 0.f, 0.f, 0.f};
#pragma unroll
        for (int kh = 0; kh < 7; ++kh) {
            const float* arow = &cur[(n + kh) * LDS_STRIDE + wo0 + 2 * khalf];
#pragma unroll
            for (int kk = 0; kk < 6; ++kk) {
                v2f a;                                       // A[m=n, k=4kk+2*khalf+j]
                a.x = arow[4 * kk];
                a.y = arow[4 * kk + 1];
                acc = __builtin_amdgcn_wmma_f32_16x16x4_f32(
                          /*neg_a=*/false, a, /*neg_b=*/false, bk[kh][kk],
                          /*c_mod=*/(short)0, acc, /*reuse_a=*/false, /*reuse_b=*/false);
            }
        }

        // ---- store 16x16 tile (D layout: lane n = col, VGPR r -> row r + 8*khalf) ----
        if (wo0 + n < WO) {
#pragma unroll
            for (int r = 0; r < 8; ++r) {
                const int ho = ho0 + r + 8 * khalf;
                if (ho < HO) op[ho * WO + wo0 + n] = acc[r];
            }
        }
    }
}

extern "C" void kernel_launch(void* const* d_in, const int* in_sizes, int n_in,
                              void* d_out, int out_size, void* d_ws, size_t ws_size,
                              hipStream_t stream) {
    const float* x = (const float*)d_in[0];
    const float* w = (const float*)d_in[1];
    float* out     = (float*)d_out;

    const int planes = in_sizes[0] / (128 * 128);   // B*C = 4096
    const int nchan  = in_sizes[1] / 49;            // 256

    dwconv7x7_wmma<<<dim3(planes), dim3(256), 0, stream>>>(x, w, out, nchan);
}